// ShortConv_36112085025011
// MI455X (gfx1250) — compile-verified
//
#include <hip/hip_runtime.h>
#include <stdint.h>

// Problem constants (fixed by the reference).
#define TT   8192
#define GG   4
#define CC   2048
#define KK   4
#define TB   16                  // output time-steps per block
#define ROWS (TB + KK - 1)       // 19 staged rows
#define C4   (CC / 4)            // 512 float4 per row
#define NTHREADS 256             // 8 wave32
#define EPSF 1e-5f

typedef float v4f __attribute__((ext_vector_type(4)));
typedef float v2f __attribute__((ext_vector_type(2)));

// gfx1250 async global->LDS path (ASYNCcnt), emitted via inline asm because
// the clang builtin's pointer params use HIP LangAS types we cannot spell.
#if defined(__gfx1250__)
#define USE_ASYNC 1
#else
#define USE_ASYNC 0
#endif

__device__ __forceinline__ void async_copy_b128(const float* gp, float* lp) {
#if USE_ASYNC
    // vdst = 32-bit LDS byte address (low 32 bits of generic LDS pointer),
    // vaddr = 64-bit global address pair.
    const uint32_t lds_addr = (uint32_t)(uintptr_t)lp;
    const uint64_t gaddr    = (uint64_t)(uintptr_t)gp;
    asm volatile("global_load_async_to_lds_b128 %0, %1, off"
                 :
                 : "v"(lds_addr), "v"(gaddr)
                 : "memory");
#else
    *(float4*)lp = *(const float4*)gp;
#endif
}

__device__ __forceinline__ void async_wait_all() {
#if USE_ASYNC
    asm volatile("s_wait_asynccnt 0" ::: "memory");
#endif
}

// Packed fp32 VALU (VOP3P): 2 FMAs / 2 MULs per instruction. ISel does not
// form these from <2 x float> IR, so emit them directly; non-volatile asm
// stays schedulable/CSE-able. 8-byte "v" operands = even-aligned VGPR pairs.
__device__ __forceinline__ v2f pk_fma(v2f a, v2f b, v2f c) {
#if defined(__gfx1250__)
    v2f d;
    asm("v_pk_fma_f32 %0, %1, %2, %3" : "=v"(d) : "v"(a), "v"(b), "v"(c));
    return d;
#else
    return a * b + c;
#endif
}

__device__ __forceinline__ v2f pk_mul(v2f a, v2f b) {
#if defined(__gfx1250__)
    v2f d;
    asm("v_pk_mul_f32 %0, %1, %2" : "=v"(d) : "v"(a), "v"(b));
    return d;
#else
    return a * b;
#endif
}

__device__ __forceinline__ float silu_f(float v) {
    // v_exp_f32 / v_rcp_f32 are TRANS-class: co-execute with the pk-FMA stream.
    return v * (1.0f / (1.0f + __expf(-v)));
}

// K=4 sliding-window conv + SiLU for one channel-quad. Weights arrive
// tap-transposed and pre-scaled by norm_weight, so the inner body is a pure
// elementwise pk-FMA chain: 8 v_pk_fma/v_pk_mul per 4 outputs.
__device__ __forceinline__ void conv_block(const float* xs, float* y,
                                           int t0, int g, int ch,
                                           v4f wt0, v4f wt1, v4f wt2, v4f wt3) {
    v4f a0 = *(const v4f*)&xs[0 * CC + ch];
    v4f a1 = *(const v4f*)&xs[1 * CC + ch];
    v4f a2 = *(const v4f*)&xs[2 * CC + ch];
#pragma unroll
    for (int tr = 0; tr < TB; ++tr) {
        const v4f a3 = *(const v4f*)&xs[(tr + 3) * CC + ch];
        v4f o;
        o.lo = pk_fma(wt0.lo, a0.lo,
               pk_fma(wt1.lo, a1.lo,
               pk_fma(wt2.lo, a2.lo, pk_mul(wt3.lo, a3.lo))));
        o.hi = pk_fma(wt0.hi, a0.hi,
               pk_fma(wt1.hi, a1.hi,
               pk_fma(wt2.hi, a2.hi, pk_mul(wt3.hi, a3.hi))));
        o.x = silu_f(o.x);
        o.y = silu_f(o.y);
        o.z = silu_f(o.z);
        o.w = silu_f(o.w);
        float* op = y + ((size_t)(t0 + tr) * GG + g) * CC + ch;
        // y is write-once, never re-read: non-temporal B128 store keeps the
        // 256 MB output stream from rinsing x/weights out of L2.
        __builtin_nontemporal_store(o, (v4f*)op);
        a0 = a1; a1 = a2; a2 = a3;           // slide the 4-row window
    }
}

extern "C" __global__ __launch_bounds__(NTHREADS)
void ShortConv_36112085025011_kernel(const float* __restrict__ x,
                                     const float* __restrict__ nw,
                                     const float* __restrict__ w,
                                     float* __restrict__ y) {
    extern __shared__ float xs[];            // ROWS * CC floats = 152 KB
    const int tid  = threadIdx.x;
    const int g    = blockIdx.y;             // group 0..3
    const int t0   = blockIdx.x * TB;        // first output time-step

    // ---- Hoisted weight loads: issue before the DMA so they overlap the ---
    // ---- staging + RMS phases (no LDS dependence). ------------------------
    const int ch_a = tid * 4;                // this thread's two channel-quads
    const int ch_b = (tid + NTHREADS) * 4;
    const int gc_a = g * CC + ch_a;
    const int gc_b = g * CC + ch_b;
    const v4f wa0 = *(const v4f*)(w + (size_t)(gc_a + 0) * KK);
    const v4f wa1 = *(const v4f*)(w + (size_t)(gc_a + 1) * KK);
    const v4f wa2 = *(const v4f*)(w + (size_t)(gc_a + 2) * KK);
    const v4f wa3 = *(const v4f*)(w + (size_t)(gc_a + 3) * KK);
    const v4f wb0 = *(const v4f*)(w + (size_t)(gc_b + 0) * KK);
    const v4f wb1 = *(const v4f*)(w + (size_t)(gc_b + 1) * KK);
    const v4f wb2 = *(const v4f*)(w + (size_t)(gc_b + 2) * KK);
    const v4f wb3 = *(const v4f*)(w + (size_t)(gc_b + 3) * KK);
    const v4f nva = *(const v4f*)(nw + gc_a);
    const v4f nvb = *(const v4f*)(nw + gc_b);

    // Tap-transpose + fold norm_weight in once: wt_k[j] = w_j[k] * nv[j].
    const v4f wta0 = (v4f){wa0.x, wa1.x, wa2.x, wa3.x} * nva;
    const v4f wta1 = (v4f){wa0.y, wa1.y, wa2.y, wa3.y} * nva;
    const v4f wta2 = (v4f){wa0.z, wa1.z, wa2.z, wa3.z} * nva;
    const v4f wta3 = (v4f){wa0.w, wa1.w, wa2.w, wa3.w} * nva;
    const v4f wtb0 = (v4f){wb0.x, wb1.x, wb2.x, wb3.x} * nvb;
    const v4f wtb1 = (v4f){wb0.y, wb1.y, wb2.y, wb3.y} * nvb;
    const v4f wtb2 = (v4f){wb0.z, wb1.z, wb2.z, wb3.z} * nvb;
    const v4f wtb3 = (v4f){wb0.w, wb1.w, wb2.w, wb3.w} * nvb;

    // ---- Phase 1: stage rows t0-3 .. t0+TB-1 of x[:, g, :] into LDS ------
    // Branch-free: clamp each row's time to >= 0 (s_max) and issue all 19x2
    // B128 async copies unconditionally; the (rare) pad rows are zero-filled
    // after the wait behind a single uniform branch.
#pragma unroll
    for (int r = 0; r < ROWS; ++r) {
        int t = t0 - (KK - 1) + r;
        t = (t < 0) ? 0 : t;                 // uniform clamp, no branch
        const float* grow = x + ((int64_t)t * GG + g) * CC;
        float* lrow = &xs[r * CC];
        async_copy_b128(grow + (size_t)ch_a, &lrow[ch_a]);
        async_copy_b128(grow + (size_t)ch_b, &lrow[ch_b]);
    }
    async_wait_all();                        // s_wait_asynccnt 0
    if (t0 == 0) {                           // only time-tile 0 pads
#pragma unroll
        for (int r = 0; r < KK - 1; ++r) {
            float* lrow = &xs[r * CC];
            *(v4f*)&lrow[ch_a] = (v4f){0.f, 0.f, 0.f, 0.f};
            *(v4f*)&lrow[ch_b] = (v4f){0.f, 0.f, 0.f, 0.f};
        }
    }
    __syncthreads();

    // ---- Phase 2: per-row RMS + in-place rescale (wave owns its rows) ----
    const int wave = tid >> 5;
    const int lane = tid & 31;
    for (int r = wave; r < ROWS; r += NTHREADS / 32) {
        float* row = &xs[r * CC];
        v2f acc_lo = (v2f){0.f, 0.f};
        v2f acc_hi = (v2f){0.f, 0.f};
        for (int i = lane; i < C4; i += 32) {
            const v4f v = *(const v4f*)&row[i * 4];
            acc_lo = pk_fma(v.lo, v.lo, acc_lo);     // packed sum of squares
            acc_hi = pk_fma(v.hi, v.hi, acc_hi);
        }
        float s = acc_lo.x + acc_lo.y + acc_hi.x + acc_hi.y;
        // wave32 butterfly: every lane ends with the full row sum
        for (int m = 16; m > 0; m >>= 1) s += __shfl_xor(s, m, 32);
        const float rms = rsqrtf(s * (1.0f / CC) + EPSF);
        const v2f rms2 = (v2f){rms, rms};
        for (int i = lane; i < C4; i += 32) {
            v4f v = *(const v4f*)&row[i * 4];
            v.lo = pk_mul(v.lo, rms2);               // packed rescale
            v.hi = pk_mul(v.hi, rms2);
            *(v4f*)&row[i * 4] = v;
        }
    }
    __syncthreads();

    // ---- Phase 3: conv + SiLU for this thread's two channel-quads --------
    conv_block(xs, y, t0, g, ch_a, wta0, wta1, wta2, wta3);
    conv_block(xs, y, t0, g, ch_b, wtb0, wtb1, wtb2, wtb3);
}

extern "C" void kernel_launch(void* const* d_in, const int* in_sizes, int n_in,
                              void* d_out, int out_size, void* d_ws, size_t ws_size,
                              hipStream_t stream) {
    (void)in_sizes; (void)n_in; (void)d_ws; (void)ws_size; (void)out_size;
    const float* x  = (const float*)d_in[0];   // (T, G, C) fp32
    const float* nw = (const float*)d_in[1];   // (G, C) fp32
    const float* w  = (const float*)d_in[2];   // (G*C, 1, K) fp32
    float* y = (float*)d_out;                  // (T, G, C) fp32

    dim3 grid(TT / TB, GG);                    // 512 x 4 blocks
    dim3 block(NTHREADS);
    const size_t shmem = (size_t)ROWS * CC * sizeof(float);   // 152 KB dyn-LDS
    ShortConv_36112085025011_kernel<<<grid, block, shmem, stream>>>(x, nw, w, y);
}